// GCN_Classic_31482110280435
// MI455X (gfx1250) — compile-verified
//
#include <hip/hip_runtime.h>
#include <hip/hip_bf16.h>

#define NN 100000
#define EE 1600000
#define DD 64
#define LL 4
#define GG 512
#define OUTC 10

typedef float v2f __attribute__((ext_vector_type(2)));
typedef float v8f __attribute__((ext_vector_type(8)));

// ---------------------------------------------------------------- init / norm
__global__ void k_init_deg(float* deg) {
    int i = blockIdx.x * blockDim.x + threadIdx.x;
    if (i < NN) deg[i] = 1.0f;
}

__global__ void k_deg_accum(const int* __restrict__ dst, float* __restrict__ deg) {
    int e = blockIdx.x * blockDim.x + threadIdx.x;   // E = 1.6M, exact grid
    atomicAdd(&deg[dst[e]], 1.0f);
}

__global__ void k_norm(float* __restrict__ deg_dis, float* __restrict__ snorm) {
    int i = blockIdx.x * blockDim.x + threadIdx.x;
    if (i < NN) {
        float d = rsqrtf(deg_dis[i]);
        deg_dis[i] = d;          // becomes dis
        snorm[i]   = d * d;
    }
}

__global__ void k_enorm(const int* __restrict__ src, const int* __restrict__ dst,
                        const float* __restrict__ dis, float* __restrict__ enorm) {
    int e = blockIdx.x * blockDim.x + threadIdx.x;   // exact grid
    enorm[e] = dis[src[e]] * dis[dst[e]];
}

// -------------------------------------------------- WMMA GEMM: [rows,64]@[64,64]
// one wave per 16x16 tile; 8 waves/block = 2 row tiles x 4 col tiles
__global__ void k_gemm64(const float* __restrict__ A, const float* __restrict__ W,
                         float* __restrict__ C) {
    __shared__ float Wl[DD * DD];                    // 16 KB weight stage
    int t = threadIdx.x;
#pragma unroll
    for (int i = 0; i < 16; ++i) Wl[t + i * 256] = W[t + i * 256];
    __syncthreads();

    int wave = t >> 5;
    int lane = t & 31;
    int half = lane >> 4;        // 0 or 1
    int l15  = lane & 15;

    int rowTile = blockIdx.x * 2 + (wave >> 2);
    int colTile = wave & 3;
    int m0 = rowTile * 16;
    int n0 = colTile * 16;

    const size_t arow = (size_t)(m0 + l15) * DD;
    v8f acc = {};
#pragma unroll
    for (int k = 0; k < DD; k += 4) {
        int ka = k + 2 * half;       // K index for VGPR0 of this lane half
        v2f a, b;
        a.x = A[arow + ka];
        a.y = A[arow + ka + 1];
        b.x = Wl[ka * DD + n0 + l15];
        b.y = Wl[(ka + 1) * DD + n0 + l15];
        acc = __builtin_amdgcn_wmma_f32_16x16x4_f32(false, a, false, b,
                                                    (short)0, acc, false, false);
    }
#pragma unroll
    for (int r = 0; r < 8; ++r)
        C[(size_t)(m0 + r + 8 * half) * DD + n0 + l15] = acc[r];
}

// ------------------------------------------------------------- edge scatter
// thread per (edge, channel): wave32 covers 32 contiguous channels -> coalesced
__global__ void k_scatter(const float* __restrict__ hw, const int* __restrict__ src,
                          const int* __restrict__ dst, const float* __restrict__ enorm,
                          float* __restrict__ agg) {
    unsigned idx = blockIdx.x * 256u + threadIdx.x;  // E*64 exact
    int e = idx >> 6;
    int c = idx & 63;
    float v = hw[(size_t)src[e] * DD + c] * enorm[e];
    atomicAdd(&agg[(size_t)dst[e] * DD + c], v);
}

// ------------------------------------------- combine + graph-pool scatter
__global__ void k_combine(const float* __restrict__ agg, const float* __restrict__ hw,
                          const float* __restrict__ snorm, const float* __restrict__ bias,
                          const int* __restrict__ batch, float* __restrict__ h,
                          float* __restrict__ pooledL) {
    unsigned idx = blockIdx.x * 256u + threadIdx.x;  // N*64 exact
    int i = idx >> 6;
    int c = idx & 63;
    float v = agg[idx] + hw[idx] * snorm[i] + bias[c];
    h[idx] = v;
    atomicAdd(&pooledL[(size_t)batch[i] * DD + c], v);
}

// -------------------------------- MLP GEMM: z1 = concat(pooled) @ W1 + b1
// M=512, K=256, N=64 -> 32x4 = 128 tiles = 16 blocks of 8 waves
__global__ void k_mlp_gemm(const float* __restrict__ pooled,   // [L][G][D]
                           const float* __restrict__ W1,       // [256][64]
                           const float* __restrict__ b1,
                           float* __restrict__ z1) {           // [G][64]
    int t = threadIdx.x;
    int wave = t >> 5, lane = t & 31, half = lane >> 4, l15 = lane & 15;
    int tile = blockIdx.x * 8 + wave;
    int m0 = (tile >> 2) * 16;
    int n0 = (tile & 3) * 16;
    int row = m0 + l15;

    v8f acc = {};
#pragma unroll 4
    for (int k = 0; k < LL * DD; k += 4) {
        int ka = k + 2 * half;
        int kb = ka + 1;
        v2f a, b;
        a.x = pooled[(size_t)(ka >> 6) * (GG * DD) + (size_t)row * DD + (ka & 63)];
        a.y = pooled[(size_t)(kb >> 6) * (GG * DD) + (size_t)row * DD + (kb & 63)];
        b.x = W1[(size_t)ka * DD + n0 + l15];
        b.y = W1[(size_t)kb * DD + n0 + l15];
        acc = __builtin_amdgcn_wmma_f32_16x16x4_f32(false, a, false, b,
                                                    (short)0, acc, false, false);
    }
#pragma unroll
    for (int r = 0; r < 8; ++r) {
        int rr = m0 + r + 8 * half;
        int cc = n0 + l15;
        z1[(size_t)rr * DD + cc] = acc[r] + b1[cc];
    }
}

// -------------------------------------------------- batch-norm statistics
__global__ void k_bn_stats(const float* __restrict__ z1, float* __restrict__ mean,
                           float* __restrict__ rstd) {
    __shared__ float s1[256], s2[256];
    int c = blockIdx.x;           // one column per block
    int t = threadIdx.x;
    float a = 0.f, b = 0.f;
    for (int g = t; g < GG; g += 256) {
        float v = z1[(size_t)g * DD + c];
        a += v; b += v * v;
    }
    s1[t] = a; s2[t] = b;
    __syncthreads();
    for (int off = 128; off > 0; off >>= 1) {
        if (t < off) { s1[t] += s1[t + off]; s2[t] += s2[t + off]; }
        __syncthreads();
    }
    if (t == 0) {
        float m = s1[0] / (float)GG;
        float var = s2[0] / (float)GG - m * m;
        mean[c] = m;
        rstd[c] = rsqrtf(var + 1e-5f);
    }
}

// ------------------------------------- normalize, relu, out linear [64->10]
__global__ void k_mlp_out(const float* __restrict__ z1, const float* __restrict__ mean,
                          const float* __restrict__ rstd, const float* __restrict__ gamma,
                          const float* __restrict__ beta, const float* __restrict__ W2,
                          const float* __restrict__ b2, float* __restrict__ out) {
    __shared__ float zn[DD];
    int g = blockIdx.x, t = threadIdx.x;
    float v = (z1[(size_t)g * DD + t] - mean[t]) * rstd[t] * gamma[t] + beta[t];
    zn[t] = v > 0.f ? v : 0.f;
    __syncthreads();
    if (t < OUTC) {
        float o = b2[t];
#pragma unroll
        for (int d = 0; d < DD; ++d) o += zn[d] * W2[d * OUTC + t];
        out[(size_t)g * OUTC + t] = o;
    }
}

// ---------------------------------------------------------------- launcher
extern "C" void kernel_launch(void* const* d_in, const int* in_sizes, int n_in,
                              void* d_out, int out_size, void* d_ws, size_t ws_size,
                              hipStream_t stream) {
    const float* x     = (const float*)d_in[0];
    const int*   ei    = (const int*)d_in[1];
    const int*   src   = ei;
    const int*   dst   = ei + EE;
    const int*   batch = (const int*)d_in[2];
    const float* Ws    = (const float*)d_in[3];
    const float* bs    = (const float*)d_in[4];
    const float* W1    = (const float*)d_in[5];
    const float* b1    = (const float*)d_in[6];
    const float* gamma = (const float*)d_in[7];
    const float* beta  = (const float*)d_in[8];
    const float* W2    = (const float*)d_in[9];
    const float* b2    = (const float*)d_in[10];
    float* out = (float*)d_out;

    float* ws     = (float*)d_ws;
    float* deg    = ws;                                   // N (becomes dis)
    float* snorm  = deg + NN;                             // N
    float* enorm  = snorm + NN;                           // E
    float* h      = enorm + EE;                           // N*D
    float* hw     = h + (size_t)NN * DD;                  // N*D
    float* agg    = hw + (size_t)NN * DD;                 // N*D
    float* pooled = agg + (size_t)NN * DD;                // L*G*D
    float* z1     = pooled + (size_t)LL * GG * DD;        // G*D
    float* mean   = z1 + (size_t)GG * DD;                 // D
    float* rstd   = mean + DD;                            // D

    // degree + normalization coefficients
    k_init_deg<<<(NN + 255) / 256, 256, 0, stream>>>(deg);
    k_deg_accum<<<EE / 256, 256, 0, stream>>>(dst, deg);
    k_norm<<<(NN + 255) / 256, 256, 0, stream>>>(deg, snorm);
    k_enorm<<<EE / 256, 256, 0, stream>>>(src, dst, deg, enorm);

    hipMemsetAsync(pooled, 0, (size_t)LL * GG * DD * sizeof(float), stream);

    for (int l = 0; l < LL; ++l) {
        const float* hin = (l == 0) ? x : h;
        k_gemm64<<<NN / 16 / 2, 256, 0, stream>>>(hin, Ws + (size_t)l * DD * DD, hw);
        hipMemsetAsync(agg, 0, (size_t)NN * DD * sizeof(float), stream);
        k_scatter<<<(EE * 64) / 256, 256, 0, stream>>>(hw, src, dst, enorm, agg);
        k_combine<<<(NN * 64) / 256, 256, 0, stream>>>(agg, hw, snorm, bs + l * DD,
                                                       batch, h, pooled + (size_t)l * GG * DD);
    }

    k_mlp_gemm<<<(GG / 16) * 4 / 8, 256, 0, stream>>>(pooled, W1, b1, z1);
    k_bn_stats<<<DD, 256, 0, stream>>>(z1, mean, rstd);
    k_mlp_out<<<GG, DD, 0, stream>>>(z1, mean, rstd, gamma, beta, W2, b2, out);
}